// TemporalRiemannianAttentionBias_8615704396370
// MI455X (gfx1250) — compile-verified
//
#include <hip/hip_runtime.h>
#include <hip/hip_bf16.h>
#include <stdint.h>

// ---------------------------------------------------------------------------
// TemporalRiemannianAttentionBias, MI455X (gfx1250, wave32, WMMA)
//
// Math: S = XX^T/d (masked), Y = I+S, L = Y^{-1}(2Y-4I) = 2I - 4*Y^{-1}.
// Y^{-1} via Newton-Schulz (Z <- Z(2I - Y Z)), Z0 = theta*I,
// theta = 2/(1+Gershgorin(Y)).  All heavy math = v_wmma_f32_16x16x4_f32.
// Ym = -Y kept in LDS (f32 WMMA has no A/B negation, only C-neg).
// 4 accumulator chains per wave (shared A operand) for WMMA ILP;
// 128-bit non-temporal loads/stores on the 288 MB streaming output.
// ---------------------------------------------------------------------------

typedef __attribute__((ext_vector_type(2))) float v2f;
typedef __attribute__((ext_vector_type(4))) float v4f;
typedef __attribute__((ext_vector_type(8))) float v8f;

#define BCN 512
#define NN  128
#define DD  256
#define HH  8
#define EPSV 1e-5f

#define XLD 268          // X row stride (floats): 256 + 12 pad  (12 mod 64 banks)
#define MLD 140          // matrix row stride:     128 + 12 pad
#define MAT_F (NN * MLD) // floats per padded 128x128 matrix
#define NS_ITERS 8
#define LDS_FLOATS (4 * MAT_F + NN + 8)

// One wave computes its 16-row block of D = A*B (+ 2*Cinit), K = 128.
// 4 column tiles in flight, sharing the A operand.
__device__ __forceinline__ void wave_mm_rowblock(
    const float* __restrict__ A, const float* __restrict__ B,
    float* __restrict__ Dm, const float* __restrict__ Cinit,
    int tr, int lo, int hi)
{
  const float* Arow = A + (tr * 16 + lo) * MLD;
  for (int tc = 0; tc < 8; tc += 4) {
    v8f acc[4] = {{}, {}, {}, {}};
    if (Cinit) {
#pragma unroll
      for (int j = 0; j < 4; ++j)
#pragma unroll
        for (int i = 0; i < 8; ++i) {
          int r = tr * 16 + i + 8 * hi;
          acc[j][i] = 2.0f * Cinit[r * MLD + (tc + j) * 16 + lo];
        }
    }
#pragma unroll 4
    for (int k = 0; k < NN; k += 4) {
      int kk = k + 2 * hi;
      v2f a;
      a.x = Arow[kk];
      a.y = Arow[kk + 1];
      const float* Br0 = B + kk * MLD;
      const float* Br1 = B + (kk + 1) * MLD;
#pragma unroll
      for (int j = 0; j < 4; ++j) {
        v2f bv;
        bv.x = Br0[(tc + j) * 16 + lo];
        bv.y = Br1[(tc + j) * 16 + lo];
        acc[j] = __builtin_amdgcn_wmma_f32_16x16x4_f32(false, a, false, bv,
                                                       (short)0, acc[j],
                                                       false, false);
      }
    }
#pragma unroll
    for (int j = 0; j < 4; ++j)
#pragma unroll
      for (int i = 0; i < 8; ++i) {
        int r = tr * 16 + i + 8 * hi;
        Dm[r * MLD + (tc + j) * 16 + lo] = acc[j][i];
      }
  }
}

__global__ void __launch_bounds__(256, 1)
trab_ns_kernel(const float* __restrict__ x,          // (512,128,256) f32
               const uint8_t* __restrict__ mask,     // (512,128) bool
               const float* __restrict__ head_scales,// (8,) f32
               const float* __restrict__ mu,         // (128,128) f32
               float* __restrict__ out_bias,         // (512,8,128,128) f32
               float* __restrict__ out_L)            // (512,128,128) f32
{
  extern __shared__ float smem[];
  float* Ym = smem;                 // -(I + S_masked)
  float* Za = smem + 1 * MAT_F;
  float* Zb = smem + 2 * MAT_F;
  float* Tm = smem + 3 * MAT_F;
  float* Xs = Zb;                   // X overlays Zb+Tm during phase 1
  float* rs = smem + 4 * MAT_F;     // mask flags -> row sums; rs[NN] = theta

  const int b    = blockIdx.x;
  const int tid  = threadIdx.x;
  const int wave = tid >> 5;
  const int lane = tid & 31;
  const int lo   = lane & 15;
  const int hi   = lane >> 4;
  const int tr   = wave;            // this wave's 16-row tile block

  // ---- Phase 0: stage X (b128 loads) and mask flags into LDS --------------
  {
    const v4f* xb4 = (const v4f*)(x + (size_t)b * (NN * DD));
    for (int q = tid; q < (NN * DD) / 4; q += 256) {
      int r  = q >> 6;              // 64 float4 per 256-wide row
      int c4 = (q & 63) << 2;
      *(v4f*)&Xs[r * XLD + c4] = __builtin_nontemporal_load(&xb4[q]);
    }
  }
  if (tid < NN) rs[tid] = mask[b * NN + tid] ? 1.0f : 0.0f;
  __syncthreads();

  // ---- Phase 1: S = X X^T / d via WMMA; build Ym = -(I + S_masked) --------
  {
    const float* Arow = Xs + (tr * 16 + lo) * XLD;
    for (int tc = 0; tc < 8; tc += 4) {
      v8f acc[4] = {{}, {}, {}, {}};
      const float* Bp[4];
#pragma unroll
      for (int j = 0; j < 4; ++j) Bp[j] = Xs + ((tc + j) * 16 + lo) * XLD;
#pragma unroll 4
      for (int k = 0; k < DD; k += 4) {
        int kk = k + 2 * hi;
        v2f a;
        a.x = Arow[kk];
        a.y = Arow[kk + 1];
#pragma unroll
        for (int j = 0; j < 4; ++j) {
          v2f bv;                        // B = X^T tile == X rows of tc+j
          bv.x = Bp[j][kk];
          bv.y = Bp[j][kk + 1];
          acc[j] = __builtin_amdgcn_wmma_f32_16x16x4_f32(false, a, false, bv,
                                                         (short)0, acc[j],
                                                         false, false);
        }
      }
#pragma unroll
      for (int i = 0; i < 8; ++i) {
        int r = tr * 16 + i + 8 * hi;
        float mr = rs[r];
#pragma unroll
        for (int j = 0; j < 4; ++j) {
          int c = (tc + j) * 16 + lo;
          float eye  = (r == c) ? 1.0f : 0.0f;
          float many = fminf(mr + rs[c], 1.0f);
          float s = acc[j][i] * (1.0f / DD) + EPSV * eye;
          s = s * (1.0f - many) + eye * many;
          Ym[r * MLD + c] = -(eye + s);
        }
      }
    }
  }
  __syncthreads();   // Ym complete; X region (Zb,Tm) now dead

  // ---- Gershgorin bound -> theta = 2/(1+lambda_max) -----------------------
  float rsum = 0.0f;
  if (tid < NN) {
    const float* row = Ym + tid * MLD;
    for (int c = 0; c < NN; ++c) rsum += fabsf(row[c]);
  }
  __syncthreads();                       // mask flags in rs[] no longer needed
  if (tid < NN) rs[tid] = rsum;
  __syncthreads();
  if (tid == 0) {
    float rmax = 0.0f;
    for (int i = 0; i < NN; ++i) rmax = fmaxf(rmax, rs[i]);
    rs[NN] = 2.0f / (1.0f + rmax);
  }
  __syncthreads();
  const float theta = rs[NN];

  // ---- Phase 2: Newton-Schulz.  Fused first step: Z1 = 2*theta*I + theta^2*Ym
  {
    float th2 = theta * theta;
    for (int idx = tid; idx < NN * NN; idx += 256) {
      int r = idx >> 7, c = idx & 127;
      Za[r * MLD + c] = th2 * Ym[r * MLD + c] + ((r == c) ? 2.0f * theta : 0.0f);
    }
  }
  __syncthreads();

  float* Zin  = Za;
  float* Zout = Zb;
  for (int it = 0; it < NS_ITERS; ++it) {
    wave_mm_rowblock(Ym, Zin, Tm, nullptr, tr, lo, hi);      // T   = -Y*Z
    __syncthreads();
    wave_mm_rowblock(Zin, Tm, Zout, Zin, tr, lo, hi);        // Z' = 2Z + Z*T
    __syncthreads();
    float* t = Zin; Zin = Zout; Zout = t;
  }
  const float* Zf = Zin;   // ~ Y^{-1}   (even #iters -> lands in Za)

  // ---- Phase 3: L = 2I - 4*sym(Z) - 0.5*(mu + mu^T); bias = L * scales ----
  // 128-bit non-temporal stores: 9 x b128 per 4 elements on the 288 MB output.
  float hs[HH];
#pragma unroll
  for (int h = 0; h < HH; ++h) hs[h] = head_scales[h];
  float* Lb = out_L + (size_t)b * (NN * NN);
  for (int idx = tid * 4; idx < NN * NN; idx += 256 * 4) {
    int r = idx >> 7, c = idx & 127;               // c..c+3 stay in row r
    v4f zrow  = *(const v4f*)&Zf[r * MLD + c];
    v4f murow = *(const v4f*)&mu[r * NN + c];
    v4f l;
#pragma unroll
    for (int j = 0; j < 4; ++j) {
      float z = 0.5f * (zrow[j] + Zf[(c + j) * MLD + r]);
      float eye = (r == (c + j)) ? 2.0f : 0.0f;
      l[j] = eye - 4.0f * z - 0.5f * (murow[j] + mu[(c + j) * NN + r]);
    }
    __builtin_nontemporal_store(l, (v4f*)&Lb[idx]);
#pragma unroll
    for (int h = 0; h < HH; ++h) {
      v4f bv = l * hs[h];
      __builtin_nontemporal_store(
          bv, (v4f*)&out_bias[((size_t)b * HH + h) * (NN * NN) + idx]);
    }
  }
}

extern "C" void kernel_launch(void* const* d_in, const int* in_sizes, int n_in,
                              void* d_out, int out_size, void* d_ws, size_t ws_size,
                              hipStream_t stream) {
  (void)in_sizes; (void)n_in; (void)out_size; (void)d_ws; (void)ws_size;
  const float*   x    = (const float*)d_in[0];
  const uint8_t* mask = (const uint8_t*)d_in[1];   // jnp.bool_ -> 1 byte/elt
  const float*   hsc  = (const float*)d_in[2];
  const float*   mu   = (const float*)d_in[3];
  float* bias = (float*)d_out;                               // (512,8,128,128)
  float* L    = bias + (size_t)BCN * HH * NN * NN;           // (512,128,128)
  size_t lds_bytes = (size_t)LDS_FLOATS * sizeof(float);     // ~280.5 KB < 320 KB
  trab_ns_kernel<<<dim3(BCN), dim3(256), lds_bytes, stream>>>(x, mask, hsc, mu,
                                                              bias, L);
}